// YGate_549755814196
// MI455X (gfx1250) — compile-verified
//
#include <hip/hip_runtime.h>

// YGate (Pauli-Y on qubit 0 of 13) applied to an 8192-element complex state.
// out[k]       = -i * psi[k+4096]
// out[k+4096]  = +i * psi[k]
// Implemented as a real 4x4 contraction per batch element k in [0,4096),
// executed on the CDNA5 matrix pipe via V_WMMA_F32_16X16X4_F32:
//   A (16x4)  = 16 batch elements x (re_lo, im_lo, re_hi, im_hi)
//   B (4x16)  = constant real expansion of Y (only 4 columns non-zero)
//   D (16x16) = columns 0..3 are the 4 output components per batch row.
// All global traffic is coalesced b128; operand marshaling goes through LDS.

typedef __attribute__((ext_vector_type(2))) float v2f;
typedef __attribute__((ext_vector_type(4))) float v4f;
typedef __attribute__((ext_vector_type(8))) float v8f;

#define NHALF 4096
#define BATCH_PER_BLOCK 1024

__global__ __launch_bounds__(256) void ygate_wmma_kernel(
    const float* __restrict__ xre, const float* __restrict__ xim,
    float* __restrict__ out)
{
  __shared__ float2 lds_lo[BATCH_PER_BLOCK];  // (re, im) of psi[k]
  __shared__ float2 lds_hi[BATCH_PER_BLOCK];  // (re, im) of psi[k+4096]

  const int t    = threadIdx.x;
  const int base = blockIdx.x * BATCH_PER_BLOCK;

  // ---- Phase 1: coalesced b128 global loads -> LDS, interleaving re/im.
  {
    const int i = t * 4;  // 0..1020, 16B aligned
    v4f rl = *(const v4f*)(xre + base + i);
    v4f il = *(const v4f*)(xim + base + i);
    v4f rh = *(const v4f*)(xre + NHALF + base + i);
    v4f ih = *(const v4f*)(xim + NHALF + base + i);
#pragma unroll
    for (int j = 0; j < 4; ++j) {
      lds_lo[i + j] = make_float2(rl[j], il[j]);
      lds_hi[i + j] = make_float2(rh[j], ih[j]);
    }
  }
  __syncthreads();

  const int lane = t & 31;   // wave32
  const int wave = t >> 5;   // 8 waves per block
  const int half = lane >> 4;
  const int mrow = lane & 15;

  // ---- B operand (constant): B[k][n], lane = n + 16*(k>=2), vgpr = k%2.
  //  B[3][0]=+1 -> lane16.v1 ; B[2][1]=-1 -> lane17.v0
  //  B[1][2]=-1 -> lane 2.v1 ; B[0][3]=+1 -> lane 3.v0
  v2f b;
  b.x = (lane == 3) ? 1.0f : ((lane == 17) ? -1.0f : 0.0f);
  b.y = (lane == 16) ? 1.0f : ((lane == 2) ? -1.0f : 0.0f);

  // Output component n -> flat offset in d_out ([2, N, 1] layout):
  //  n=0: out_re[k]      -> 0      n=1: out_im[k]      -> 8192
  //  n=2: out_re[k+4096] -> 4096   n=3: out_im[k+4096] -> 12288
  const int n = lane & 15;
  const int off_n = (n & 1) * 8192 + ((n >> 1) & 1) * 4096;

  // ---- Phase 2: 8 WMMA tiles of 16 batch elements per wave.
#pragma unroll
  for (int tile = 0; tile < 8; ++tile) {
    const int tb = wave * 128 + tile * 16;  // batch tile base within block
    // A operand: lanes 0-15 = (re,im) of psi[k]; lanes 16-31 = of psi[k+4096]
    float2 p = half ? lds_hi[tb + mrow] : lds_lo[tb + mrow];
    v2f a;
    a.x = p.x;
    a.y = p.y;
    v8f c = {};
    v8f d = __builtin_amdgcn_wmma_f32_16x16x4_f32(
        false, a, false, b, (short)0, c, false, false);
    // D layout: lane (n + 16*hd), VGPR r holds D[r + 8*hd][n].
    // Active lanes (n<4) hold 8 consecutive batch rows -> two b128 stores.
    if (n < 4) {
      float* dst = out + off_n + base + tb + 8 * half;  // 32B aligned
      v4f dlo = { d[0], d[1], d[2], d[3] };
      v4f dhi = { d[4], d[5], d[6], d[7] };
      *(v4f*)(dst)     = dlo;
      *(v4f*)(dst + 4) = dhi;
    }
  }
}

extern "C" void kernel_launch(void* const* d_in, const int* in_sizes, int n_in,
                              void* d_out, int out_size, void* d_ws, size_t ws_size,
                              hipStream_t stream) {
  const float* xre = (const float*)d_in[0];  // 8192 floats
  const float* xim = (const float*)d_in[1];  // 8192 floats
  float* out = (float*)d_out;                // 16384 floats: [re | im]
  dim3 grid(NHALF / BATCH_PER_BLOCK);        // 4 blocks
  dim3 block(256);                           // 8 wave32s
  hipLaunchKernelGGL(ygate_wmma_kernel, grid, block, 0, stream, xre, xim, out);
}